// MGEmbedding_558345748968
// MI455X (gfx1250) — compile-verified
//
#include <hip/hip_runtime.h>

// MI455X / gfx1250, wave32. fp32 GEMM via V_WMMA_F32_16X16X4_F32 with fused FiLM.

typedef __attribute__((ext_vector_type(2))) float v2f;
typedef __attribute__((ext_vector_type(8))) float v8f;

#define FDIM   128
#define NCELLS 12288
#define NHS    9      // adjc row stride
#define BM     64     // rows of h per workgroup
#define BN     256    // cols of h per workgroup (scale|shift of one child)
#define A_STR  132    // 128 + 4 -> conflict-free b64 A-frag reads
#define BT_STR 132    // transposed B tile: Bt[n][k], conflict-free b64 reads
#define H_STR  260    // 256 + 4 -> conflict-free accumulator spill

__launch_bounds__(256)
__global__ void mg_film_gemm(
    const float* __restrict__ emb,     // (NVARS, NCELLS, FDIM)
    const float* __restrict__ W,       // (nhk*FDIM, cols) row-major
    const float* __restrict__ bias,    // (cols)
    const float* __restrict__ xin,     // (8, childFactor*NCELLS, FDIM)
    float*       __restrict__ yout,    // same shape as xin
    const int*   __restrict__ var_idx, // (8)
    const int*   __restrict__ adjc,    // (NCELLS, NHS)
    int nhk, int cols, int childFactor)
{
    extern __shared__ float smem[];
    float* Alds = smem;                          // BM x A_STR
    float* Blds = smem + BM * A_STR;             // BN x BT_STR (transposed)
    int*   ridx = (int*)(smem + BM * A_STR + BN * BT_STR); // BM ints
    float* Hlds = smem;                          // BM x H_STR (reuses A/B space)

    const int t    = threadIdx.x;
    const int lane = t & 31;
    const int wave = t >> 5;

    // 2 row-strips x 4 col-strips per wave: 8 waves tile 64x256
    const int rowPair = wave & 1;       // strips {2rp, 2rp+1}
    const int colQuad = wave >> 1;      // col strips 4cq .. 4cq+3

    const int mBase = blockIdx.x * BM;
    const int child = blockIdx.y;
    const int cBase = child * BN;
    const int bv    = blockIdx.z;               // flat (b,t,v)
    const int var   = var_idx[bv];
    const float* embV = emb + (size_t)var * NCELLS * FDIM;

    v8f acc[2][4];
    #pragma unroll
    for (int r = 0; r < 2; ++r)
        #pragma unroll
        for (int s = 0; s < 4; ++s)
            #pragma unroll
            for (int p = 0; p < 8; ++p) acc[r][s][p] = 0.0f;

    const int koff  = (lane < 16) ? 0 : 2;
    const int arow0 = (rowPair * 2) * 16 + (lane & 15);   // first of 2 strips
    const int bcol0 = colQuad * 64 + (lane & 15);

    for (int j = 0; j < nhk; ++j) {
        __syncthreads();   // previous inner loop / first entry
        if (t < BM) ridx[t] = adjc[(size_t)(mBase + t) * NHS + j];
        __syncthreads();

        // --- stage A: 64 gathered embedding rows, coalesced b128 ---
        {
            const int fv = (t & 31) * 4;
            #pragma unroll
            for (int r0 = 0; r0 < BM; r0 += 8) {
                const int r = r0 + (t >> 5);
                const float4 v =
                    *(const float4*)(embV + (size_t)ridx[r] * FDIM + fv);
                *(float4*)(Alds + r * A_STR + fv) = v;
            }
        }
        // --- stage B transposed: thread t owns column n=t, walks k ---
        {
            const float* wp = W + (size_t)(j * FDIM) * cols + cBase + t;
            float* bt = Blds + t * BT_STR;
            #pragma unroll 4
            for (int k = 0; k < FDIM; k += 4) {
                float4 v;
                v.x = wp[(size_t)(k + 0) * cols];
                v.y = wp[(size_t)(k + 1) * cols];
                v.z = wp[(size_t)(k + 2) * cols];
                v.w = wp[(size_t)(k + 3) * cols];
                *(float4*)(bt + k) = v;   // ds_store_b128, banks 4n+k..+3
            }
        }
        __syncthreads();

        // --- inner product: 32 k-steps, 6 b64 LDS loads feed 8 WMMAs ---
        #pragma unroll 4
        for (int k2 = 0; k2 < FDIM; k2 += 4) {
            const v2f a0 = *(const v2f*)(Alds + arow0 * A_STR + k2 + koff);
            const v2f a1 = *(const v2f*)(Alds + (arow0 + 16) * A_STR + k2 + koff);
            #pragma unroll
            for (int s = 0; s < 4; ++s) {
                const v2f b =
                    *(const v2f*)(Blds + (bcol0 + s * 16) * BT_STR + k2 + koff);
                acc[0][s] = __builtin_amdgcn_wmma_f32_16x16x4_f32(
                    false, a0, false, b, (short)0, acc[0][s], false, false);
                acc[1][s] = __builtin_amdgcn_wmma_f32_16x16x4_f32(
                    false, a1, false, b, (short)0, acc[1][s], false, false);
            }
        }
    }
    __syncthreads();

    // --- spill accumulators (f32 C/D layout: VGPR p -> rows p / p+8) ---
    #pragma unroll
    for (int r = 0; r < 2; ++r) {
        const int rbase = (rowPair * 2 + r) * 16 + ((lane < 16) ? 0 : 8);
        #pragma unroll
        for (int s = 0; s < 4; ++s) {
            const int cc = colQuad * 64 + s * 16 + (lane & 15);
            #pragma unroll
            for (int p = 0; p < 8; ++p)
                Hlds[(rbase + p) * H_STR + cc] = acc[r][s][p];
        }
    }
    __syncthreads();

    // --- fused bias + FiLM epilogue, coalesced x/y ---
    const size_t ntot = (size_t)childFactor * NCELLS;
    for (int idx = t; idx < BM * FDIM; idx += 256) {
        const int r = idx >> 7;
        const int f = idx & (FDIM - 1);
        const float sc = Hlds[r * H_STR + f]        + bias[cBase + f];
        const float sh = Hlds[r * H_STR + FDIM + f] + bias[cBase + FDIM + f];
        const size_t u = (size_t)childFactor * (mBase + r) + child;
        const size_t o = ((size_t)bv * ntot + u) * FDIM + f;
        yout[o] = xin[o] * sc + sh;
    }
}

extern "C" void kernel_launch(void* const* d_in, const int* in_sizes, int n_in,
                              void* d_out, int out_size, void* d_ws, size_t ws_size,
                              hipStream_t stream) {
    const float* emb = (const float*)d_in[0];
    const float* x5  = (const float*)d_in[1];
    const float* x6  = (const float*)d_in[2];
    const float* W5  = (const float*)d_in[3];
    const float* b5  = (const float*)d_in[4];
    const float* W6  = (const float*)d_in[5];
    const float* b6  = (const float*)d_in[6];
    const int* var_idx = (const int*)d_in[7];
    const int* adjc    = (const int*)d_in[8];

    float* y5 = (float*)d_out;
    float* y6 = y5 + (size_t)8 * NCELLS * FDIM;   // y5 is (8, N, F)

    const size_t smemBytes =
        (size_t)(BM * A_STR + BN * BT_STR + BM) * sizeof(float); // ~169 KB

    dim3 blk(256);
    // Stage 5: K = 128 (j=0 only), cols = 256, child factor 1
    mg_film_gemm<<<dim3(NCELLS / BM, 1, 8), blk, smemBytes, stream>>>(
        emb, W5, b5, x5, y5, var_idx, adjc, 1, 256, 1);
    // Stage 6: K = 9*128, cols = 1024, 4 children per coarse cell
    mg_film_gemm<<<dim3(NCELLS / BM, 4, 8), blk, smemBytes, stream>>>(
        emb, W6, b6, x6, y6, var_idx, adjc, 9, 1024, 4);
}